// GRUDecoder_249108103972
// MI455X (gfx1250) — compile-verified
//
#include <hip/hip_runtime.h>
#include <math.h>

// ---------------------------------------------------------------------------
// Problem constants (from reference): S,B,N,H,DIN,DOUT,T = 12,64,500,64,2,1,12
// ---------------------------------------------------------------------------
#define SS   12
#define BB   64
#define NN   500
#define HH   64
#define DIN  2
#define TT   12
#define NH   (NN * HH)        // 32000
#define ROWS (BB * NN)        // 32000   rows of the hidden state
#define KROWS (BB * TT * NN)  // 384000  rows of encoder outputs / k

typedef float v2f __attribute__((ext_vector_type(2)));
typedef float v8f __attribute__((ext_vector_type(8)));

// D = A(16x4) * B(4x16) + C  in fp32 on the matrix pipe
__device__ __forceinline__ v8f wmma_f32_4(v2f a, v2f b, v8f c) {
  return __builtin_amdgcn_wmma_f32_16x16x4_f32(
      /*neg_a=*/false, a, /*neg_b=*/false, b,
      /*c_mod=*/(short)0, c, /*reuse_a=*/false, /*reuse_b=*/false);
}

// A fragment: A row-major [M x lda], tile rows m0..m0+15, k-step ks (4 per step)
__device__ __forceinline__ v2f load_a_g(const float* __restrict__ A, int lda,
                                        int m0, int ks, int lane) {
  int m  = m0 + (lane & 15);
  int kb = 4 * ks + 2 * (lane >> 4);
  const float* p = A + m * lda + kb;
  v2f a; a.x = p[0]; a.y = p[1]; return a;
}

// B fragment from weight W row-major [Nout x ldw]; computes B[k][n] = W[n][koff+k]
__device__ __forceinline__ v2f load_b_w(const float* __restrict__ W, int ldw,
                                        int n0, int koff, int ks, int lane) {
  int n  = n0 + (lane & 15);
  int kb = koff + 4 * ks + 2 * (lane >> 4);
  const float* p = W + n * ldw + kb;
  v2f b; b.x = p[0]; b.y = p[1]; return b;
}

// ---------------------------------------------------------------------------
// C[M x 64] = A[M x 64] @ W[64 x 64]^T + bias   (used for k-proj and q-proj)
// grid.x = M/16 blocks, 128 threads = 4 waves, wave w owns n-tile w.
// ---------------------------------------------------------------------------
__global__ __launch_bounds__(128) void gemm64_kernel(
    const float* __restrict__ A, const float* __restrict__ W,
    const float* __restrict__ bias, float* __restrict__ C) {
  int lane = threadIdx.x & 31;
  int wave = threadIdx.x >> 5;
  int m0 = blockIdx.x * 16;
  int n0 = wave * 16;
  v8f acc = {};
  for (int ks = 0; ks < 16; ++ks) {
    v2f a = load_a_g(A, 64, m0, ks, lane);
    v2f b = load_b_w(W, 64, n0, 0, ks, lane);
    acc = wmma_f32_4(a, b, acc);
  }
  int n = n0 + (lane & 15);
  float bz = bias[n];
  int mrow = m0 + 8 * (lane >> 4);
#pragma unroll
  for (int i = 0; i < 8; ++i) C[(mrow + i) * 64 + n] = acc[i] + bz;
}

// ---------------------------------------------------------------------------
// energy[b,t] = (1/sqrt(N*H)) * dot(q[b,:], k[b,t,:,:])   (both contiguous 32000)
// grid.x = B*T, 256 threads, LDS tree reduction (deterministic).
// ---------------------------------------------------------------------------
__global__ __launch_bounds__(256) void energy_kernel(
    const float* __restrict__ q, const float* __restrict__ k,
    float* __restrict__ energy, float rscale) {
  int b = blockIdx.x / TT, t = blockIdx.x % TT;
  const float* qp = q + b * NH;
  const float* kp = k + (size_t)(b * TT + t) * NH;
  float s = 0.f;
  for (int d = threadIdx.x; d < NH; d += 256) s += qp[d] * kp[d];
  __shared__ float red[256];
  red[threadIdx.x] = s;
  __syncthreads();
  for (int off = 128; off > 0; off >>= 1) {
    if (threadIdx.x < off) red[threadIdx.x] += red[threadIdx.x + off];
    __syncthreads();
  }
  if (threadIdx.x == 0) energy[blockIdx.x] = red[0] * rscale;
}

// softmax over T=12 per batch element; 1 block of 64 threads (one per b)
__global__ __launch_bounds__(64) void softmax_kernel(
    const float* __restrict__ energy, float* __restrict__ attn) {
  int b = threadIdx.x;
  float e[TT];
  float m = -3.4e38f;
#pragma unroll
  for (int t = 0; t < TT; ++t) { e[t] = energy[b * TT + t]; m = fmaxf(m, e[t]); }
  float s = 0.f;
#pragma unroll
  for (int t = 0; t < TT; ++t) { e[t] = __expf(e[t] - m); s += e[t]; }
  float inv = 1.f / s;
#pragma unroll
  for (int t = 0; t < TT; ++t) attn[b * TT + t] = e[t] * inv;
}

// ---------------------------------------------------------------------------
// weighted[b,d] = sum_t attn[b,t] * v[b,t,d] with the reference's scrambled
// reshape: v[b,t,d] = kT_flat[b][t*NH+d], kT[b,i,j] = k[b, j, i/H, i%H].
// ---------------------------------------------------------------------------
__global__ __launch_bounds__(256) void weighted_kernel(
    const float* __restrict__ attn, const float* __restrict__ k,
    float* __restrict__ weighted) {
  int gid = blockIdx.x * 256 + threadIdx.x;  // < B*NH
  int b = gid / NH;
  int d = gid - b * NH;
  const float* at = attn + b * TT;
  float acc = 0.f;
#pragma unroll
  for (int t2 = 0; t2 < TT; ++t2) {
    int e  = t2 * NH + d;        // < 12*32000, fits int
    int t  = e % TT;
    int r  = e / TT;
    int n2 = r >> 6;             // r / H
    int h2 = r & 63;             // r % H
    acc += at[t2] * k[(size_t)((b * TT + t) * NN + n2) * HH + h2];
  }
  weighted[gid] = acc;
}

// ---------------------------------------------------------------------------
// Fused GRU cell + output head for one step.
// Block = 128 threads (4 waves) handles 16 rows of the 32000-row state.
//   Stage 1: gi = weighted @ w_ih[:,2:66]^T ; gh = h @ w_hh^T   (WMMA, K=64)
//   Stage 2: add biases + Din=2 x-part, gates, h_new -> LDS + global
//   Stage 3: o1 = relu(h_new @ p1_w^T + p1_b)                   (WMMA)
//   Stage 4: out = o1 @ p2_w^T + p2_b (64 -> 1 matvec)
// ---------------------------------------------------------------------------
__global__ __launch_bounds__(128) void gru_kernel(
    const float* __restrict__ weighted, const float* __restrict__ h_in,
    const float* __restrict__ x,      // inputs for this step, [ROWS x 2]
    const float* __restrict__ w_ih,   // [192 x 66]
    const float* __restrict__ w_hh,   // [192 x 64]
    const float* __restrict__ b_ih, const float* __restrict__ b_hh,
    const float* __restrict__ p1_w,   // [64 x 64]
    const float* __restrict__ p1_b,
    const float* __restrict__ p2_w,   // [1 x 64]
    const float* __restrict__ p2_b,
    float* __restrict__ h_out, float* __restrict__ out_step) {
  __shared__ float gi[16 * 200];   // 16 rows x 192 (padded)
  __shared__ float gh[16 * 200];
  __shared__ float hn[16 * 66];    // h_new tile (padded for bank spread)
  __shared__ float o1[16 * 66];

  int lane = threadIdx.x & 31;
  int wave = threadIdx.x >> 5;
  int m0 = blockIdx.x * 16;

  // ---- Stage 1: 24 WMMA tile-tasks (12 gi + 12 gh), 6 per wave ----
  for (int task = wave; task < 24; task += 4) {
    bool isGi = task < 12;
    int nt = isGi ? task : task - 12;
    const float* A = isGi ? weighted : h_in;
    const float* W = isGi ? w_ih : w_hh;
    int ldw  = isGi ? 66 : 64;
    int koff = isGi ? 2 : 0;     // skip the Din=2 columns of w_ih
    v8f acc = {};
    for (int ks = 0; ks < 16; ++ks) {
      v2f a = load_a_g(A, 64, m0, ks, lane);
      v2f b = load_b_w(W, ldw, nt * 16, koff, ks, lane);
      acc = wmma_f32_4(a, b, acc);
    }
    float* dst = isGi ? gi : gh;
    int n = nt * 16 + (lane & 15);
    int rbase = 8 * (lane >> 4);
#pragma unroll
    for (int i = 0; i < 8; ++i) dst[(rbase + i) * 200 + n] = acc[i];
  }
  __syncthreads();

  // ---- Stage 2: biases + x-part + gates ----
  for (int idx = threadIdx.x; idx < 16 * 64; idx += 128) {
    int r = idx >> 6;
    int j = idx & 63;
    int row = m0 + r;
    float x0 = x[row * 2 + 0];
    float x1 = x[row * 2 + 1];
    float gir = gi[r * 200 + j]       + b_ih[j]       + x0 * w_ih[(j)       * 66] + x1 * w_ih[(j)       * 66 + 1];
    float giz = gi[r * 200 + 64 + j]  + b_ih[64 + j]  + x0 * w_ih[(64 + j)  * 66] + x1 * w_ih[(64 + j)  * 66 + 1];
    float gin = gi[r * 200 + 128 + j] + b_ih[128 + j] + x0 * w_ih[(128 + j) * 66] + x1 * w_ih[(128 + j) * 66 + 1];
    float ghr = gh[r * 200 + j]       + b_hh[j];
    float ghz = gh[r * 200 + 64 + j]  + b_hh[64 + j];
    float ghn = gh[r * 200 + 128 + j] + b_hh[128 + j];
    float rg = 1.f / (1.f + __expf(-(gir + ghr)));
    float zg = 1.f / (1.f + __expf(-(giz + ghz)));
    float ng = tanhf(gin + rg * ghn);
    float hprev = h_in[row * 64 + j];
    float hnew = (1.f - zg) * ng + zg * hprev;
    hn[r * 66 + j] = hnew;
    h_out[row * 64 + j] = hnew;
  }
  __syncthreads();

  // ---- Stage 3: o1 = relu(h_new @ p1_w^T + p1_b), wave w -> n-tile w ----
  {
    v8f acc = {};
    for (int ks = 0; ks < 16; ++ks) {
      int m  = lane & 15;
      int kb = 4 * ks + 2 * (lane >> 4);
      v2f a; a.x = hn[m * 66 + kb]; a.y = hn[m * 66 + kb + 1];
      v2f b = load_b_w(p1_w, 64, wave * 16, 0, ks, lane);
      acc = wmma_f32_4(a, b, acc);
    }
    int n = wave * 16 + (lane & 15);
    float bz = p1_b[n];
    int rbase = 8 * (lane >> 4);
#pragma unroll
    for (int i = 0; i < 8; ++i) {
      float v = acc[i] + bz;
      o1[(rbase + i) * 66 + n] = v > 0.f ? v : 0.f;
    }
  }
  __syncthreads();

  // ---- Stage 4: out = o1 @ p2_w^T + p2_b ----
  if (threadIdx.x < 16) {
    int r = threadIdx.x;
    float s = p2_b[0];
    for (int j = 0; j < 64; ++j) s += o1[r * 66 + j] * p2_w[j];
    out_step[m0 + r] = s;   // row = b*500 + n == flat [b, n] index
  }
}

// ---------------------------------------------------------------------------
// Host-side launch sequence (deterministic; all on `stream`).
// ---------------------------------------------------------------------------
extern "C" void kernel_launch(void* const* d_in, const int* in_sizes, int n_in,
                              void* d_out, int out_size, void* d_ws, size_t ws_size,
                              hipStream_t stream) {
  const float* inputs = (const float*)d_in[0];   // [S, B, N, 2]
  const float* h_init = (const float*)d_in[1];   // [B, N, H]
  const float* enc    = (const float*)d_in[2];   // [B, T, N, H]
  const float* Wq_w   = (const float*)d_in[3];
  const float* Wq_b   = (const float*)d_in[4];
  const float* Wk_w   = (const float*)d_in[5];
  const float* Wk_b   = (const float*)d_in[6];
  const float* w_ih   = (const float*)d_in[7];   // [192, 66]
  const float* w_hh   = (const float*)d_in[8];   // [192, 64]
  const float* b_ih   = (const float*)d_in[9];
  const float* b_hh   = (const float*)d_in[10];
  const float* p1_w   = (const float*)d_in[11];
  const float* p1_b   = (const float*)d_in[12];
  const float* p2_w   = (const float*)d_in[13];
  const float* p2_b   = (const float*)d_in[14];
  float* out = (float*)d_out;

  // Workspace layout (fp32)
  float* ws   = (float*)d_ws;
  float* kArr = ws;                          // KROWS*64 = 24,576,000
  float* h0   = kArr + (size_t)KROWS * 64;   // ROWS*64
  float* h1   = h0 + (size_t)ROWS * 64;
  float* q    = h1 + (size_t)ROWS * 64;
  float* wtd  = q + (size_t)ROWS * 64;
  float* egy  = wtd + (size_t)ROWS * 64;     // B*T
  float* attn = egy + BB * TT;               // B*T

  const float rscale = 1.0f / sqrtf((float)(NN * HH));

  // One-time key projection: k = enc @ Wk_w^T + Wk_b  (L2-resident afterwards)
  gemm64_kernel<<<KROWS / 16, 128, 0, stream>>>(enc, Wk_w, Wk_b, kArr);

  // h <- initial hidden state
  hipMemcpyAsync(h0, h_init, (size_t)ROWS * 64 * sizeof(float),
                 hipMemcpyDeviceToDevice, stream);

  float* hcur = h0;
  float* hnxt = h1;
  for (int s = 0; s < SS - 1; ++s) {
    gemm64_kernel<<<ROWS / 16, 128, 0, stream>>>(hcur, Wq_w, Wq_b, q);
    energy_kernel<<<BB * TT, 256, 0, stream>>>(q, kArr, egy, rscale);
    softmax_kernel<<<1, 64, 0, stream>>>(egy, attn);
    weighted_kernel<<<(BB * NH) / 256, 256, 0, stream>>>(attn, kArr, wtd);
    gru_kernel<<<ROWS / 16, 128, 0, stream>>>(
        wtd, hcur, inputs + (size_t)s * ROWS * DIN,
        w_ih, w_hh, b_ih, b_hh, p1_w, p1_b, p2_w, p2_b,
        hnxt, out + (size_t)s * BB * NN);
    float* t = hcur; hcur = hnxt; hnxt = t;
  }
}